// InfGaussMMVAE_52020643889860
// MI455X (gfx1250) — compile-verified
//
#include <hip/hip_runtime.h>
#include <hip/hip_bf16.h>

#define TPB   256
#define EPT   8                 // elements per thread
#define TILE  (TPB * EPT)       // 2048 elements per workgroup
#define WSLICE 256              // 32 lanes * 8 elements per wave

typedef unsigned int u32x4 __attribute__((ext_vector_type(4)));
typedef int          i32x8 __attribute__((ext_vector_type(8)));
typedef int          i32x4 __attribute__((ext_vector_type(4)));

// ---------------------------------------------------------------------------
// TDM: 1-D tensor tile (global -> LDS), data_size = 4 bytes.
// tileElems rows are read; reads past tensorElems return zero (OOB fill).
// ---------------------------------------------------------------------------
__device__ __forceinline__ void tdm_load_1d_f32(const float* gptr,
                                                unsigned ldsByteOff,
                                                int tensorElems,
                                                int tileElems) {
#if defined(__gfx1250__)
  unsigned long long ga = (unsigned long long)(const void*)gptr;
  unsigned td = (tensorElems > 0) ? (unsigned)tensorElems : 0u;

  u32x4 g0;
  g0.x = 1u;                                            // count=1 (valid user D#)
  g0.y = ldsByteOff;                                    // lds_addr
  g0.z = (unsigned)(ga & 0xFFFFFFFFu);                  // global_addr[31:0]
  g0.w = (unsigned)((ga >> 32) & 0x01FFFFFFu) | (2u << 30); // global_addr[56:32], type=2

  i32x8 g1;
  g1[0] = (int)(2u << 16);                              // wg_mask=0, data_size=2 (4B)
  g1[1] = (int)((td & 0xFFFFu) << 16);                  // abar_addr=0, tensor_dim0[15:0]
  g1[2] = (int)((td >> 16) | (1u << 16));               // tensor_dim0[31:16], tensor_dim1=1
  g1[3] = (int)(((unsigned)tileElems & 0xFFFFu) << 16); // tensor_dim1[31:16]=0, tile_dim0
  g1[4] = 1;                                            // tile_dim1=1, tile_dim2=0
  g1[5] = tileElems;                                    // tensor_dim0_stride[31:0]
  g1[6] = 0;
  g1[7] = 0;

  i32x4 z4 = {0, 0, 0, 0};
  i32x8 z8 = {0, 0, 0, 0, 0, 0, 0, 0};
  __builtin_amdgcn_tensor_load_to_lds(g0, g1, z4, z4, z8, 0);
#else
  (void)gptr; (void)ldsByteOff; (void)tensorElems; (void)tileElems;
#endif
}

// ---------------------------------------------------------------------------
// Compact lgamma(x) for 0 < x < ~32 (covers m/a <= 22, x+b <= 24 here).
// Shift x up to >= 8 accumulating the product, then Stirling.
// Cold path only (series is skipped when beta == 1).
// ---------------------------------------------------------------------------
__device__ __forceinline__ float lgammaf_pos(float x) {
  float p = 1.0f;
#pragma unroll
  for (int i = 0; i < 8; ++i) {
    float lt = (x < 8.0f) ? 1.0f : 0.0f;
    p  = p * (lt != 0.0f ? x : 1.0f);
    x += lt;
  }
  float xi  = 1.0f / x;
  float xi2 = xi * xi;
  float tail = xi * (0.0833333333f - xi2 * (0.0027777778f - xi2 * 0.00079365079f));
  return (x - 0.5f) * logf(x) - x + 0.91893853320467274f + tail - logf(p);
}

// ---------------------------------------------------------------------------
// digamma(x) for x > 0 (here x = b + eps in [0.5, 2)).
// Unconditional 6-step recurrence with pairwise-combined reciprocals
// (3 divides), then asymptotic series at y = x + 6 >= 6.5:
//   psi(x) = ln y - 1/(2y) - 1/(12y^2) + 1/(120y^4) - 1/(252y^6) - S
//   S = (2x+1)/(x(x+1)) + (2x+5)/((x+2)(x+3)) + (2x+9)/((x+4)(x+5))
// ---------------------------------------------------------------------------
__device__ __forceinline__ float digammaf_dev(float x) {
  float s0 = (2.0f * x + 1.0f) / (x * (x + 1.0f));
  float s1 = (2.0f * x + 5.0f) / ((x + 2.0f) * (x + 3.0f));
  float s2 = (2.0f * x + 9.0f) / ((x + 4.0f) * (x + 5.0f));
  float y   = x + 6.0f;
  float yi  = 1.0f / y;
  float yi2 = yi * yi;
  float tail = yi2 * (0.083333333333f - yi2 * (0.008333333333f - yi2 * 0.003968253968f));
  return logf(y) - 0.5f * yi - tail - (s0 + s1 + s2);
}

__device__ __forceinline__ float kl_elem(float a, float b, float alpha, float beta,
                                         float logBetaAB, bool needSeries) {
  const float SMALL = 1e-16f;
  const float EG    = 0.57721566490153286f;

  float ab   = fmaf(a, b, SMALL);
  float ainv = 1.0f / (a + SMALL);
  float binv = 1.0f / (b + SMALL);

  float kl = 0.0f;
  if (needSeries) {                       // wave-uniform; exactly 0 when beta == 1
    float lgb = lgammaf_pos(b);
    float acc = 0.0f;
#pragma unroll 1
    for (int i = 1; i <= 11; ++i) {
      float m  = (float)i;
      float x  = m * ainv;
      float lg = lgammaf_pos(x) + lgb - lgammaf_pos(x + b);
      acc += expf(lg) / (m + ab);
    }
    kl = (beta - 1.0f) * b * acc;
  }

  kl += ((a - alpha) * ainv) * (-EG - digammaf_dev(b + SMALL) - binv);
  kl += logf(ab) + logBetaAB;
  kl -= (b - 1.0f) * binv;
  return kl;
}

// ---------------------------------------------------------------------------
// Main elementwise kernel: per-wave TDM staging into LDS, vectorized compute.
// ---------------------------------------------------------------------------
__global__ __launch_bounds__(TPB) void kum_kl_kernel(const float* __restrict__ a,
                                                     const float* __restrict__ b,
                                                     const float* __restrict__ prm,
                                                     float* __restrict__ out,
                                                     int n) {
  __shared__ float sA[TILE];
  __shared__ float sB[TILE];

  const int wave = threadIdx.x >> 5;
  const int lane = threadIdx.x & 31;
  const long long blockBase = (long long)blockIdx.x * TILE;
  const int  waveOff  = wave * WSLICE;
  const long long waveBase = blockBase + (long long)waveOff;

  long long remainLL = (long long)n - waveBase;
  int tensorElems = (remainLL >= WSLICE) ? WSLICE : (remainLL > 0 ? (int)remainLL : 0);

#if defined(__gfx1250__)
  {
    unsigned ldsA = (unsigned)(unsigned long long)(const void*)(sA + waveOff);
    unsigned ldsB = (unsigned)(unsigned long long)(const void*)(sB + waveOff);
    tdm_load_1d_f32(a + waveBase, ldsA, tensorElems, WSLICE);
    tdm_load_1d_f32(b + waveBase, ldsB, tensorElems, WSLICE);
  }
#endif

  const float logBetaAB = prm[0];
  const float alpha     = prm[1];
  const float beta      = prm[2];
  const bool  needSeries = (beta != 1.0f);

#if defined(__gfx1250__)
  __builtin_amdgcn_s_wait_tensorcnt(0);
#else
  // Portable fallback (host pass / non-gfx1250): plain staged copy.
  for (int i = threadIdx.x; i < TILE; i += TPB) {
    long long g = blockBase + i;
    sA[i] = (g < n) ? a[g] : 0.0f;
    sB[i] = (g < n) ? b[g] : 0.0f;
  }
  __syncthreads();
#endif

  if (blockBase + TILE <= (long long)n) {
    // Full tile: 2 x float4 per thread, per-wave slice, ds_load_b128 path.
#pragma unroll
    for (int k = 0; k < 2; ++k) {
      int v = lane + 32 * k;   // float4 index inside this wave's 256-elem slice
      float4 va = ((const float4*)(sA + waveOff))[v];
      float4 vb = ((const float4*)(sB + waveOff))[v];
      float4 r;
      r.x = kl_elem(va.x, vb.x, alpha, beta, logBetaAB, needSeries);
      r.y = kl_elem(va.y, vb.y, alpha, beta, logBetaAB, needSeries);
      r.z = kl_elem(va.z, vb.z, alpha, beta, logBetaAB, needSeries);
      r.w = kl_elem(va.w, vb.w, alpha, beta, logBetaAB, needSeries);
      ((float4*)(out + waveBase))[v] = r;
    }
  } else {
    // Tail tile: scalar, bounds-checked (cold; keep un-unrolled for size).
#pragma unroll 1
    for (int k = 0; k < EPT; ++k) {
      int e = lane + 32 * k;
      long long g = waveBase + e;
      if (g < n) {
        out[g] = kl_elem(sA[waveOff + e], sB[waveOff + e],
                         alpha, beta, logBetaAB, needSeries);
      }
    }
  }
}

// ---------------------------------------------------------------------------
// 1-thread prep: scalar log(Beta(alpha,beta)+eps) and the raw scalars.
// Uses libm lgammaf for full-range correctness (runs once; size irrelevant).
// ---------------------------------------------------------------------------
__global__ void kum_kl_prep(const float* __restrict__ alpha,
                            const float* __restrict__ beta,
                            float* __restrict__ prm) {
  float al = alpha[0], be = beta[0];
  float lb = lgammaf(al) + lgammaf(be) - lgammaf(al + be);
  prm[0] = logf(expf(lb) + 1e-16f);
  prm[1] = al;
  prm[2] = be;
}

// ---------------------------------------------------------------------------
extern "C" void kernel_launch(void* const* d_in, const int* in_sizes, int n_in,
                              void* d_out, int out_size, void* d_ws, size_t ws_size,
                              hipStream_t stream) {
  const float* a     = (const float*)d_in[0];
  const float* b     = (const float*)d_in[1];
  const float* alpha = (const float*)d_in[2];
  const float* beta  = (const float*)d_in[3];
  float* out = (float*)d_out;
  float* prm = (float*)d_ws;

  int n = in_sizes[0];

  kum_kl_prep<<<1, 1, 0, stream>>>(alpha, beta, prm);

  int grid = (n + TILE - 1) / TILE;
  kum_kl_kernel<<<grid, TPB, 0, stream>>>(a, b, prm, out, n);
}